// GNN_21706764714518
// MI455X (gfx1250) — compile-verified
//
#include <hip/hip_runtime.h>
#include <hip/hip_bf16.h>

// ---------------------------------------------------------------------------
// GATv2 x3 pipeline for MI455X (gfx1250, wave32).
// Dense projections use V_WMMA_F32_16X16X4_F32 (full fp32 tensor op).
// Edge softmax/scatter uses global f32 atomics (L2-resident working set).
// ---------------------------------------------------------------------------

#define NNODES 50000
#define NEDGES 400000
#define E2     (NEDGES + NNODES)   // with self loops
#define FHID   256                 // H*C
#define NEG_SLOPE 0.2f
#define BN_EPS 1e-5f

#define CDIV(a,b) (((a)+(b)-1)/(b))

typedef float v2f __attribute__((ext_vector_type(2)));
typedef float v8f __attribute__((ext_vector_type(8)));

// ---- helpers --------------------------------------------------------------

__device__ inline void atomicMaxFloat(float* addr, float v) {
  // sign-aware trick: ints order like floats for >=0; unsigned order reverses for <0
  if (v >= 0.f) atomicMax((int*)addr, __float_as_int(v));
  else          atomicMin((unsigned int*)addr, __float_as_uint(v));
}

__device__ inline float leaky(float v) { return v > 0.f ? v : NEG_SLOPE * v; }

// ---- small utility kernels ------------------------------------------------

__global__ void k_fill(float* p, float v, int n) {
  int i = blockIdx.x * blockDim.x + threadIdx.x;
  if (i < n) p[i] = v;
}

__global__ void k_bias_init(float* out, const float* __restrict__ bias, int total, int C) {
  int i = blockIdx.x * blockDim.x + threadIdx.x;
  if (i < total) out[i] = bias[i % C];
}

// h0 = concat(x, known) : [N,20]
__global__ void k_build_h0(const float* __restrict__ x, const unsigned char* __restrict__ known,
                           float* __restrict__ h0, int N) {
  int i = blockIdx.x * blockDim.x + threadIdx.x;
  if (i >= N * 20) return;
  int n = i / 20, c = i % 20;
  h0[i] = (c < 10) ? x[n * 10 + c] : (known[n * 10 + (c - 10)] ? 1.f : 0.f);
}

// self-loop attrs: mean of incoming edge attrs (fill_value='mean')
__global__ void k_loop_accum(const int* __restrict__ ei, const float* __restrict__ eattr,
                             float* __restrict__ sums, float* __restrict__ cnt, int E) {
  int e = blockIdx.x * blockDim.x + threadIdx.x;
  if (e >= E) return;
  int dst = ei[E + e];
  atomicAdd(&cnt[dst], 1.f);
  atomicAdd(&sums[dst * 3 + 0], eattr[e * 3 + 0]);
  atomicAdd(&sums[dst * 3 + 1], eattr[e * 3 + 1]);
  atomicAdd(&sums[dst * 3 + 2], eattr[e * 3 + 2]);
}

__global__ void k_loop_final(float* __restrict__ mean_ea, const float* __restrict__ cnt, int N) {
  int n = blockIdx.x * blockDim.x + threadIdx.x;
  if (n >= N) return;
  float c = fmaxf(cnt[n], 1.f);
  mean_ea[n * 3 + 0] /= c;
  mean_ea[n * 3 + 1] /= c;
  mean_ea[n * 3 + 2] /= c;
}

// ---- FP32 WMMA GEMM: C[MxNc] = A[MxK] @ W[KxNc] + bias -------------------
// M % 16 == 0, Nc % 16 == 0, K % 4 == 0. One 16x16 tile per wave32.
// Uses V_WMMA_F32_16X16X4_F32 (A: 16x4 f32 in 2 VGPRs, B: 4x16 in 2 VGPRs,
// C/D: 8 VGPRs; lanes 0-15 rows M=r, lanes 16-31 rows M=r+8).
__global__ void k_gemm_wmma_f32(const float* __restrict__ A, const float* __restrict__ W,
                                const float* __restrict__ bias, float* __restrict__ C,
                                int M, int K, int Nc) {
  int wavesPerBlock = blockDim.x >> 5;
  int wid  = blockIdx.x * wavesPerBlock + (threadIdx.x >> 5);
  int lane = threadIdx.x & 31;
  int tilesN = Nc >> 4;
  int tm = wid / tilesN;
  int tn = wid - tm * tilesN;
  if (tm * 16 >= M) return;                 // wave-uniform

  int mn   = lane & 15;                     // M (for A) / N (for B,C)
  int half = (lane >> 4);                   // 0: K{0,1}, 1: K{2,3}
  int m = tm * 16 + mn;
  int n = tn * 16 + mn;

  v8f acc = {};
  for (int k = 0; k < K; k += 4) {
    v2f a, b;
    int ka = k + 2 * half;
    a[0] = A[m * K + ka];
    a[1] = A[m * K + ka + 1];
    b[0] = W[ka * Nc + n];
    b[1] = W[(ka + 1) * Nc + n];
    acc = __builtin_amdgcn_wmma_f32_16x16x4_f32(
        /*neg_a=*/false, a, /*neg_b=*/false, b,
        /*c_mod=*/(short)0, acc, /*reuse_a=*/false, /*reuse_b=*/false);
  }

  int rowbase = tm * 16 + (half << 3);
  float bi = bias[n];
#pragma unroll
  for (int r = 0; r < 8; ++r)
    C[(rowbase + r) * Nc + n] = acc[r] + bi;
}

// ---- edge passes (H=4, C=64), one wave32 per edge -------------------------

__device__ inline void get_edge(int e, const int* __restrict__ ei,
                                const float* __restrict__ eattr,
                                const float* __restrict__ mean_ea,
                                int& src, int& dst, float& a0, float& a1, float& a2) {
  if (e < NEDGES) {
    src = ei[e]; dst = ei[NEDGES + e];
    a0 = eattr[e * 3]; a1 = eattr[e * 3 + 1]; a2 = eattr[e * 3 + 2];
  } else {
    src = dst = e - NEDGES;
    const float* m = mean_ea + src * 3;
    a0 = m[0]; a1 = m[1]; a2 = m[2];
  }
}

// pass A: logits e[E2,4] + per-(dst,head) running max
__global__ void k_edge_logits(const int* __restrict__ ei, const float* __restrict__ eattr,
                              const float* __restrict__ mean_ea,
                              const float* __restrict__ xl, const float* __restrict__ xr,
                              const float* __restrict__ we, const float* __restrict__ att,
                              float* __restrict__ ebuf, float* __restrict__ mx) {
  int lane = threadIdx.x & 31;
  int e = blockIdx.x * (blockDim.x >> 5) + (threadIdx.x >> 5);
  if (e >= E2) return;                      // wave-uniform
  int src, dst; float a0, a1, a2;
  get_edge(e, ei, eattr, mean_ea, src, dst, a0, a1, a2);
  const float* xls = xl + src * FHID;
  const float* xrd = xr + dst * FHID;
  int c0 = lane * 8;                        // lanes 0-7 -> head0, ... 24-31 -> head3
  float partial = 0.f;
#pragma unroll
  for (int j = 0; j < 8; ++j) {
    int ch = c0 + j;
    float v = xls[ch] + xrd[ch] + a0 * we[ch] + a1 * we[256 + ch] + a2 * we[512 + ch];
    partial += att[ch] * leaky(v);          // att flat [H*C]: head*64 + c == ch
  }
  partial += __shfl_xor(partial, 1);
  partial += __shfl_xor(partial, 2);
  partial += __shfl_xor(partial, 4);
  if ((lane & 7) == 0) {
    int head = lane >> 3;
    ebuf[e * 4 + head] = partial;
    atomicMaxFloat(&mx[dst * 4 + head], partial);
  }
}

// pass B: ex = exp(e - max); segment sum
__global__ void k_edge_expsum(const int* __restrict__ ei, float* __restrict__ ebuf,
                              const float* __restrict__ mx, float* __restrict__ sm) {
  int i = blockIdx.x * blockDim.x + threadIdx.x;
  if (i >= E2 * 4) return;
  int e = i >> 2, h = i & 3;
  int dst = (e < NEDGES) ? ei[NEDGES + e] : (e - NEDGES);
  float ex = __expf(ebuf[i] - mx[dst * 4 + h]);
  ebuf[i] = ex;
  atomicAdd(&sm[dst * 4 + h], ex);
}

// pass C: out[dst] += alpha * xl[src]
__global__ void k_edge_scatter(const int* __restrict__ ei, const float* __restrict__ eattr,
                               const float* __restrict__ mean_ea,
                               const float* __restrict__ xl, const float* __restrict__ ebuf,
                               const float* __restrict__ sm, float* __restrict__ out) {
  int lane = threadIdx.x & 31;
  int e = blockIdx.x * (blockDim.x >> 5) + (threadIdx.x >> 5);
  if (e >= E2) return;
  int src, dst; float a0, a1, a2;
  get_edge(e, ei, eattr, mean_ea, src, dst, a0, a1, a2);
  int head = lane >> 3;
  float alpha = ebuf[e * 4 + head] / (sm[dst * 4 + head] + 1e-16f);
  const float* xls = xl + src * FHID;
  float* od = out + dst * FHID;
  int c0 = lane * 8;
#pragma unroll
  for (int j = 0; j < 8; ++j)
    atomicAdd(&od[c0 + j], alpha * xls[c0 + j]);
}

// ---- BatchNorm (training stats over node dim) + ELU -----------------------

__global__ void k_bn_accum(const float* __restrict__ X, float* __restrict__ s,
                           float* __restrict__ q, int N) {
  int ch = threadIdx.x;                     // 256 channels
  int row0 = blockIdx.x * 128;
  float ls = 0.f, lq = 0.f;
  for (int r = 0; r < 128; ++r) {
    int n = row0 + r;
    if (n < N) { float v = X[n * FHID + ch]; ls += v; lq += v * v; }
  }
  atomicAdd(&s[ch], ls);
  atomicAdd(&q[ch], lq);
}

__global__ void k_bn_final(const float* __restrict__ s, const float* __restrict__ q,
                           float* __restrict__ mean, float* __restrict__ inv, int N) {
  int ch = threadIdx.x;
  float m = s[ch] / (float)N;
  float var = q[ch] / (float)N - m * m;     // biased variance (matches reference)
  mean[ch] = m;
  inv[ch]  = rsqrtf(var + BN_EPS);
}

__global__ void k_bn_apply_elu(float* __restrict__ X, const float* __restrict__ mean,
                               const float* __restrict__ inv, const float* __restrict__ g,
                               const float* __restrict__ b, int total) {
  int i = blockIdx.x * blockDim.x + threadIdx.x;
  if (i >= total) return;
  int ch = i & 255;
  float v = (X[i] - mean[ch]) * inv[ch] * g[ch] + b[ch];
  X[i] = v > 0.f ? v : (__expf(v) - 1.f);   // ELU
}

// ---- layer 3 (fin=276 via virtual concat [h2 | h0], H=C=1) ---------------

__global__ void k_node3(const float* __restrict__ h2, const float* __restrict__ h0,
                        const float* __restrict__ wl, const float* __restrict__ wr,
                        const float* __restrict__ bl, const float* __restrict__ br,
                        float* __restrict__ xl3, float* __restrict__ xr3, int N) {
  int n = blockIdx.x * blockDim.x + threadIdx.x;
  if (n >= N) return;
  float sl = bl[0], sr = br[0];
  const float* a = h2 + n * FHID;
  for (int k = 0; k < FHID; ++k) { sl += a[k] * wl[k]; sr += a[k] * wr[k]; }
  const float* c = h0 + n * 20;
  for (int k = 0; k < 20; ++k) { sl += c[k] * wl[FHID + k]; sr += c[k] * wr[FHID + k]; }
  xl3[n] = sl; xr3[n] = sr;
}

__global__ void k_edge3_logits(const int* __restrict__ ei, const float* __restrict__ eattr,
                               const float* __restrict__ mean_ea,
                               const float* __restrict__ xl3, const float* __restrict__ xr3,
                               const float* __restrict__ we, const float* __restrict__ att,
                               float* __restrict__ e3, float* __restrict__ mx3) {
  int e = blockIdx.x * blockDim.x + threadIdx.x;
  if (e >= E2) return;
  int src, dst; float a0, a1, a2;
  get_edge(e, ei, eattr, mean_ea, src, dst, a0, a1, a2);
  float v = xl3[src] + xr3[dst] + a0 * we[0] + a1 * we[1] + a2 * we[2];
  float lg = att[0] * leaky(v);
  e3[e] = lg;
  atomicMaxFloat(&mx3[dst], lg);
}

__global__ void k_edge3_expsum(const int* __restrict__ ei, float* __restrict__ e3,
                               const float* __restrict__ mx3, float* __restrict__ sm3) {
  int e = blockIdx.x * blockDim.x + threadIdx.x;
  if (e >= E2) return;
  int dst = (e < NEDGES) ? ei[NEDGES + e] : (e - NEDGES);
  float ex = __expf(e3[e] - mx3[dst]);
  e3[e] = ex;
  atomicAdd(&sm3[dst], ex);
}

__global__ void k_edge3_scatter(const int* __restrict__ ei, const float* __restrict__ e3,
                                const float* __restrict__ sm3, const float* __restrict__ xl3,
                                float* __restrict__ out) {
  int e = blockIdx.x * blockDim.x + threadIdx.x;
  if (e >= E2) return;
  int src = (e < NEDGES) ? ei[e] : (e - NEDGES);
  int dst = (e < NEDGES) ? ei[NEDGES + e] : (e - NEDGES);
  atomicAdd(&out[dst], e3[e] / (sm3[dst] + 1e-16f) * xl3[src]);
}

// ---------------------------------------------------------------------------
// Launch. d_in follows jax tree-flatten order of the setup_inputs() dict:
// top-level insertion order {x, known, edge_index, edge_attr, params}; nested
// dicts in sorted-key order: params -> bn1{b,g}, bn2{b,g},
// l1{att,bl,bo,br,we,wl,wr}, l2{...}, l3{...}.
// ---------------------------------------------------------------------------
extern "C" void kernel_launch(void* const* d_in, const int* in_sizes, int n_in,
                              void* d_out, int out_size, void* d_ws, size_t ws_size,
                              hipStream_t stream) {
  const float*         x     = (const float*)d_in[0];
  const unsigned char* known = (const unsigned char*)d_in[1];
  const int*           ei    = (const int*)d_in[2];
  const float*         eattr = (const float*)d_in[3];
  const float *bn1_b = (const float*)d_in[4],  *bn1_g = (const float*)d_in[5];
  const float *bn2_b = (const float*)d_in[6],  *bn2_g = (const float*)d_in[7];
  const float *att1 = (const float*)d_in[8],  *bl1 = (const float*)d_in[9],
              *bo1  = (const float*)d_in[10], *br1 = (const float*)d_in[11],
              *we1  = (const float*)d_in[12], *wl1 = (const float*)d_in[13],
              *wr1  = (const float*)d_in[14];
  const float *att2 = (const float*)d_in[15], *bl2 = (const float*)d_in[16],
              *bo2  = (const float*)d_in[17], *br2 = (const float*)d_in[18],
              *we2  = (const float*)d_in[19], *wl2 = (const float*)d_in[20],
              *wr2  = (const float*)d_in[21];
  const float *att3 = (const float*)d_in[22], *bl3 = (const float*)d_in[23],
              *bo3  = (const float*)d_in[24], *br3 = (const float*)d_in[25],
              *we3  = (const float*)d_in[26], *wl3 = (const float*)d_in[27],
              *wr3  = (const float*)d_in[28];
  float* out = (float*)d_out;                 // [N,1] f32

  const int N = NNODES;
  // workspace arena (floats)
  float* p = (float*)d_ws;
  size_t off = 0;
  auto alloc = [&](size_t nf) { float* r = p + off; off += nf; return r; };
  float* h0      = alloc((size_t)N * 20);
  float* mean_ea = alloc((size_t)N * 3);
  float* cnt     = alloc((size_t)N);
  float* xl      = alloc((size_t)N * FHID);
  float* xr      = alloc((size_t)N * FHID);
  float* bufA    = alloc((size_t)N * FHID);   // layer1 out -> h1
  float* bufB    = alloc((size_t)N * FHID);   // layer2 out -> h2
  float* ebuf    = alloc((size_t)E2 * 4);
  float* mx      = alloc((size_t)N * 4);
  float* sm      = alloc((size_t)N * 4);
  float* st_s    = alloc(256);
  float* st_q    = alloc(256);
  float* st_m    = alloc(256);
  float* st_i    = alloc(256);
  float* xl3     = alloc((size_t)N);
  float* xr3     = alloc((size_t)N);
  float* e3      = alloc((size_t)E2);
  float* mx3     = alloc((size_t)N);
  float* sm3     = alloc((size_t)N);
  (void)ws_size;

  const int T = 256;
  const int edgeWaveBlocks = CDIV(E2, 8);     // 8 waves (=8 edges) per 256-thread block
  const int gemmBlocks = ((N / 16) * (FHID / 16)) / 4; // 4 waves/block, exact tiling

  // 0) inputs -> h0 ; self-loop mean attrs
  k_build_h0<<<CDIV(N * 20, T), T, 0, stream>>>(x, known, h0, N);
  k_fill<<<CDIV(N * 3, T), T, 0, stream>>>(mean_ea, 0.f, N * 3);
  k_fill<<<CDIV(N, T), T, 0, stream>>>(cnt, 0.f, N);
  k_loop_accum<<<CDIV(NEDGES, T), T, 0, stream>>>(ei, eattr, mean_ea, cnt, NEDGES);
  k_loop_final<<<CDIV(N, T), T, 0, stream>>>(mean_ea, cnt, N);

  // ---- layer 1 (fin=20) ----
  k_gemm_wmma_f32<<<gemmBlocks, 128, 0, stream>>>(h0, wl1, bl1, xl, N, 20, FHID);
  k_gemm_wmma_f32<<<gemmBlocks, 128, 0, stream>>>(h0, wr1, br1, xr, N, 20, FHID);
  k_fill<<<CDIV(N * 4, T), T, 0, stream>>>(mx, -3.0e38f, N * 4);
  k_fill<<<CDIV(N * 4, T), T, 0, stream>>>(sm, 0.f, N * 4);
  k_bias_init<<<CDIV(N * FHID, T), T, 0, stream>>>(bufA, bo1, N * FHID, FHID);
  k_edge_logits<<<edgeWaveBlocks, T, 0, stream>>>(ei, eattr, mean_ea, xl, xr, we1, att1, ebuf, mx);
  k_edge_expsum<<<CDIV(E2 * 4, T), T, 0, stream>>>(ei, ebuf, mx, sm);
  k_edge_scatter<<<edgeWaveBlocks, T, 0, stream>>>(ei, eattr, mean_ea, xl, ebuf, sm, bufA);
  // BN1 + ELU (in place)
  k_fill<<<1, T, 0, stream>>>(st_s, 0.f, 256);
  k_fill<<<1, T, 0, stream>>>(st_q, 0.f, 256);
  k_bn_accum<<<CDIV(N, 128), 256, 0, stream>>>(bufA, st_s, st_q, N);
  k_bn_final<<<1, 256, 0, stream>>>(st_s, st_q, st_m, st_i, N);
  k_bn_apply_elu<<<CDIV(N * FHID, T), T, 0, stream>>>(bufA, st_m, st_i, bn1_g, bn1_b, N * FHID);

  // ---- layer 2 (fin=256) ----
  k_gemm_wmma_f32<<<gemmBlocks, 128, 0, stream>>>(bufA, wl2, bl2, xl, N, 256, FHID);
  k_gemm_wmma_f32<<<gemmBlocks, 128, 0, stream>>>(bufA, wr2, br2, xr, N, 256, FHID);
  k_fill<<<CDIV(N * 4, T), T, 0, stream>>>(mx, -3.0e38f, N * 4);
  k_fill<<<CDIV(N * 4, T), T, 0, stream>>>(sm, 0.f, N * 4);
  k_bias_init<<<CDIV(N * FHID, T), T, 0, stream>>>(bufB, bo2, N * FHID, FHID);
  k_edge_logits<<<edgeWaveBlocks, T, 0, stream>>>(ei, eattr, mean_ea, xl, xr, we2, att2, ebuf, mx);
  k_edge_expsum<<<CDIV(E2 * 4, T), T, 0, stream>>>(ei, ebuf, mx, sm);
  k_edge_scatter<<<edgeWaveBlocks, T, 0, stream>>>(ei, eattr, mean_ea, xl, ebuf, sm, bufB);
  // BN2 + ELU (in place)
  k_fill<<<1, T, 0, stream>>>(st_s, 0.f, 256);
  k_fill<<<1, T, 0, stream>>>(st_q, 0.f, 256);
  k_bn_accum<<<CDIV(N, 128), 256, 0, stream>>>(bufB, st_s, st_q, N);
  k_bn_final<<<1, 256, 0, stream>>>(st_s, st_q, st_m, st_i, N);
  k_bn_apply_elu<<<CDIV(N * FHID, T), T, 0, stream>>>(bufB, st_m, st_i, bn2_g, bn2_b, N * FHID);

  // ---- layer 3 (fin=276 via [h2|h0], out=1) ----
  k_node3<<<CDIV(N, T), T, 0, stream>>>(bufB, h0, wl3, wr3, bl3, br3, xl3, xr3, N);
  k_fill<<<CDIV(N, T), T, 0, stream>>>(mx3, -3.0e38f, N);
  k_fill<<<CDIV(N, T), T, 0, stream>>>(sm3, 0.f, N);
  k_bias_init<<<CDIV(N, T), T, 0, stream>>>(out, bo3, N, 1);
  k_edge3_logits<<<CDIV(E2, T), T, 0, stream>>>(ei, eattr, mean_ea, xl3, xr3, we3, att3, e3, mx3);
  k_edge3_expsum<<<CDIV(E2, T), T, 0, stream>>>(ei, e3, mx3, sm3);
  k_edge3_scatter<<<CDIV(E2, T), T, 0, stream>>>(ei, e3, sm3, xl3, out);
}